// Q_Conv_Layer_83623013253700
// MI455X (gfx1250) — compile-verified
//
#include <hip/hip_runtime.h>

// ---------------------------------------------------------------------------
// Quantum 2x2-patch conv layer, collapsed to:
//   per patch:  [Re psi; Im psi] (32) = W(32x32) * [cos(phi_b); sin(phi_b)] (32)
//   then probs = Re^2+Im^2 and 4 signed 16-term reductions (CNOT ring folded
//   into compile-time sign masks).
// GEMM now uses V_WMMA_F32_16X16X4_F32 (8 K-chunks x 2 tiles = 16 WMMAs per
// 16 patches) -> exact fp32, no f16 splits, and the f32 B layout lets each
// lane compute only 8 phases. Kernel is VALU-bound (HBM floor ~1.1us), so the
// design minimizes per-group VALU: ~180 ops vs ~500 in the f16-split version.
// ---------------------------------------------------------------------------

typedef __attribute__((ext_vector_type(2))) float v2f;
typedef __attribute__((ext_vector_type(8))) float v8f;

#define ROT_HALF 0.3925f /* 0.785 * 0.5 */

// qubit 0 is the most-significant bit of the flattened 16-state index
__device__ __host__ constexpr int bitq(int b, int q) { return (b >> (3 - q)) & 1; }
__device__ __host__ constexpr int cnot_m(int b, int c, int t) {
  return b ^ (bitq(b, c) << (3 - t));
}
// source index in the pre-CNOT-ring state for measured index f:
// state4[f] = state0[ m01(m12(m23(m30(f)))) ]
__device__ __host__ constexpr int gmap(int f) {
  int b = cnot_m(f, 3, 0);
  b = cnot_m(b, 2, 3);
  b = cnot_m(b, 1, 2);
  b = cnot_m(b, 0, 1);
  return b;
}
__device__ __host__ constexpr int ginv(int a) {
  for (int f = 0; f < 16; ++f)
    if (gmap(f) == a) return f;
  return 0;
}
// lane-local basis indices (q2 bit excluded): e -> {0,1,4,5,8,9,12,13}
__device__ __host__ constexpr int blv(int e) { return ((e >> 1) << 2) | (e & 1); }

__global__ __launch_bounds__(256)
void qconv_wmma(const float* __restrict__ img, const float* __restrict__ wts,
                float* __restrict__ out, int ngroups, int npatch) {
  const int lane    = threadIdx.x & 31;
  const bool hiHalf = lane >= 16;   // selects K pairs {4c+2,4c+3} and D rows M>=8
  const int m       = lane & 15;    // A row / patch column within tile
  const int wavesPerBlk = blockDim.x >> 5;
  const int wave    = blockIdx.x * wavesPerBlk + (threadIdx.x >> 5);
  const int nwaves  = gridDim.x * wavesPerBlk;

  const float sgn2f = hiHalf ?  1.0f : -1.0f;  // sigma(q2 bit) for this half
  const float shalf = hiHalf ? -1.0f :  1.0f;  // epilogue half-dependent sign

  // ---- fixed matrix W = (1/4)[[ReR,-ImR],[ImR,ReR]], R = RX0x..xRX3 --------
  float cw[4], sw[4];
#pragma unroll
  for (int q = 0; q < 4; ++q) {
    float w = wts[q] * 0.5f;
    cw[q] = __builtin_cosf(w);
    sw[q] = __builtin_sinf(w);
  }
  // f32 16x4 A layout: lanes0-15 hold (K=4c,4c+1), lanes16-31 hold (K=4c+2,4c+3)
  const int koff = hiHalf ? 2 : 0;
  v2f A0[8], A1[8];  // tile0 rows -> Re(psi), tile1 rows -> Im(psi)
#pragma unroll
  for (int c = 0; c < 8; ++c) {
#pragma unroll
    for (int j = 0; j < 2; ++j) {
      const int k = 4 * c + koff + j;       // column 0..31 of W
      const bool sincol = (c >= 4);         // columns >=16 multiply sin part
      const int colb = k & 15;              // basis index of column
      const int xb = m ^ colb;              // mismatching qubits
      float prod = 0.25f;
#pragma unroll
      for (int q = 0; q < 4; ++q)
        prod *= ((xb >> (3 - q)) & 1) ? sw[q] : cw[q];
      const int d = __popc((unsigned)xb);   // (-i)^d
      float re = (d == 0 || d == 4) ? prod : ((d == 2) ? -prod : 0.0f);
      float im = (d == 1) ? -prod : ((d == 3) ? prod : 0.0f);
      A0[c][j] = sincol ? -im : re;
      A1[c][j] = sincol ?  re : im;
    }
  }

  for (int grp = wave; grp < ngroups; grp += nwaves) {  // uniform per wave
    // ---- load one 2x2 patch per column (both lane halves load same patch) --
    int p  = grp * 16 + m;
    int pc = p < npatch ? p : npatch - 1;
    int bi = pc / 196;
    int rr = pc - bi * 196;
    int j  = rr / 14;
    int k  = rr - j * 14;
    const float* src = img + bi * 784 + j * 56 + k * 2;  // 8B-aligned
    float2 r0 = *(const float2*)(src);
    float2 r1 = *(const float2*)(src + 28);
    float x0 = r0.x, x1 = r0.y, x2 = r1.x, x3 = r1.y;

    float t0 = x0 * ROT_HALF, t1 = x1 * ROT_HALF;
    float t2 = x2 * ROT_HALF, t3 = x3 * ROT_HALF;
    float p01 = x1 * t0, p02 = x2 * t0, p03 = x3 * t0;
    float p12 = x2 * t1, p13 = x3 * t1, p23 = x3 * t2;

    // ---- 8 phases per lane (q2 bit supplied by lane half via one FMA) ------
    float cv[8], sv[8];
#pragma unroll
    for (int e = 0; e < 8; ++e) {
      const int b  = blv(e);               // bits of q0,q1,q3
      const int s0 = bitq(b, 0), s1 = bitq(b, 1), s3 = bitq(b, 3);
      float base = (s0 ? t0 : -t0) + (s1 ? t1 : -t1) + (s3 ? t3 : -t3);
      base += (s0 == s1) ? -p01 : p01;     // -sigma_i*sigma_j * p_ij
      base += (s0 == s3) ? -p03 : p03;
      base += (s1 == s3) ? -p13 : p13;
      float q2p = t2;                      // sigma2 factored out
      q2p += s0 ? -p02 : p02;
      q2p += s1 ? -p12 : p12;
      q2p += s3 ? -p23 : p23;
      float phi = __builtin_fmaf(sgn2f, q2p, base);
      cv[e] = __cosf(phi);
      sv[e] = __sinf(phi);
    }

    // ---- fp32 GEMM: 8 K-chunks of 4, two 16-row tiles  (16 WMMAs) ----------
    v8f accR = {};
    v8f accI = {};
#pragma unroll
    for (int c = 0; c < 8; ++c) {
      const int e0 = 2 * (c & 3);
      v2f B;
      B[0] = (c < 4) ? cv[e0]     : sv[e0];
      B[1] = (c < 4) ? cv[e0 + 1] : sv[e0 + 1];
      accR = __builtin_amdgcn_wmma_f32_16x16x4_f32(false, A0[c], false, B,
                                                   (short)0, accR, false, false);
      accI = __builtin_amdgcn_wmma_f32_16x16x4_f32(false, A1[c], false, B,
                                                   (short)0, accI, false, false);
    }

    // ---- probabilities + signed reductions (all signs compile-time) --------
    float pr[8], mpr[8];
#pragma unroll
    for (int r = 0; r < 8; ++r) {
      pr[r]  = accR[r] * accR[r] + accI[r] * accI[r];
      mpr[r] = shalf * pr[r];   // DCE'd where unused
    }
    float ez[4];
#pragma unroll
    for (int q = 0; q < 4; ++q) {
      float s = 0.0f;
#pragma unroll
      for (int r = 0; r < 8; ++r) {
        const int fL = ginv(r);        // measured index if low half
        const int fH = ginv(r + 8);    // measured index if high half
        const int bL = bitq(fL, q), bH = bitq(fH, q);
        const float v = (bL == bH) ? pr[r] : mpr[r];
        s += bL ? -v : v;
      }
      ez[q] = s;
    }
    // combine the two lane halves (amps 0-7 with amps 8-15 of same patch)
    ez[0] += __shfl_xor(ez[0], 16, 32);
    ez[1] += __shfl_xor(ez[1], 16, 32);
    ez[2] += __shfl_xor(ez[2], 16, 32);
    ez[3] += __shfl_xor(ez[3], 16, 32);

    if (!hiHalf && p < npatch) {
      // out (B,4,14,14); channel order [Z3, Z2, Z1, Z0]
      float* o = out + bi * 784 + j * 14 + k;
      o[0]   = ez[3];
      o[196] = ez[2];
      o[392] = ez[1];
      o[588] = ez[0];
    }
  }
}

extern "C" void kernel_launch(void* const* d_in, const int* in_sizes, int n_in,
                              void* d_out, int out_size, void* d_ws, size_t ws_size,
                              hipStream_t stream) {
  (void)n_in; (void)out_size; (void)d_ws; (void)ws_size;
  const float* img = (const float*)d_in[0];   // (B,1,28,28) fp32
  const float* wts = (const float*)d_in[1];   // (1,4) fp32
  float* out = (float*)d_out;                 // (B,4,14,14) fp32

  int B = in_sizes[0] / 784;
  int npatch  = B * 196;
  int ngroups = (npatch + 15) / 16;           // 16 patches per wave-iteration
  const int wavesPerBlk = 8;                  // 256 threads = 8 wave32
  const int groupsPerWave = 8;                // amortize W-matrix setup
  int blocks = (ngroups + wavesPerBlk * groupsPerWave - 1) /
               (wavesPerBlk * groupsPerWave);
  if (blocks < 1) blocks = 1;
  qconv_wmma<<<blocks, 256, 0, stream>>>(img, wts, out, ngroups, npatch);
}